// GCN_Encoder_62105227100333
// MI455X (gfx1250) — compile-verified
//
#include <hip/hip_runtime.h>
#include <cstdint>

#define N_NODES 100000
#define IN_C    64
#define HID_C   128
#define OUT_C   64
#define N_EDGES 1600000

typedef __attribute__((ext_vector_type(2))) float v2f;
typedef __attribute__((ext_vector_type(8))) float v8f;

// ---------------------------------------------------------------- utilities
__global__ void zero_kernel(float* __restrict__ p, long long n) {
    long long i = (long long)blockIdx.x * blockDim.x + threadIdx.x;
    long long stride = (long long)gridDim.x * blockDim.x;
    for (; i < n; i += stride) p[i] = 0.0f;
}

// deg[col[e]] += 1  (float segment_sum of ones, matches reference)
__global__ void degree_kernel(const long long* __restrict__ col,
                              float* __restrict__ deg, int E) {
    int e = blockIdx.x * blockDim.x + threadIdx.x;
    if (e < E) atomicAdd(&deg[(int)col[e]], 1.0f);
}

// dis = deg > 0 ? rsqrt(deg) : 0   (deg is an integer >= 1 when > 0)
__global__ void rsqrt_kernel(float* __restrict__ deg, int n) {
    int i = blockIdx.x * blockDim.x + threadIdx.x;
    if (i < n) {
        float d = deg[i];
        deg[i] = (d > 0.0f) ? rsqrtf(d) : 0.0f;
    }
}

// ------------------------------------------------- fp32 WMMA GEMM  C = A*B
// A: [M,K] row-major, B: [K,N] row-major, C: [M,N] row-major.
// One wave computes a 16x16 tile of C using V_WMMA_F32_16X16X4_F32.
// Launch: grid.x = M/16 tiles, blockDim.x = (N/16)*32 (one wave per N-tile).
// VGPR layouts per CDNA5 ISA 7.12.2 (32-bit A 16x4 / B 4x16 / C 16x16).
__global__ void gemm_wmma_f32(const float* __restrict__ A,
                              const float* __restrict__ B,
                              float* __restrict__ C,
                              int M, int K, int N) {
    const int lane = threadIdx.x & 31;
    const int wave = threadIdx.x >> 5;
    const int l    = lane & 15;   // row/col within 16
    const int half = lane >> 4;   // 0: lanes 0-15, 1: lanes 16-31
    const int m0   = blockIdx.x * 16;
    const int n0   = wave * 16;

    int mA = m0 + l;
    if (mA >= M) mA = M - 1;            // clamp (M is a multiple of 16 here)
    const float* Arow = A + (long long)mA * K;

    v8f acc = {};
    for (int k = 0; k < K; k += 4) {
        // A 16x4: VGPR0 = K = k+2*half, VGPR1 = K = k+2*half+1
        v2f a;
        a.x = Arow[k + 2 * half];
        a.y = Arow[k + 2 * half + 1];
        // B 4x16: same K striping across lane halves, N = n0 + l
        const float* Bp = B + (long long)(k + 2 * half) * N + n0 + l;
        v2f b;
        b.x = Bp[0];
        b.y = Bp[N];
        acc = __builtin_amdgcn_wmma_f32_16x16x4_f32(
            /*neg_a=*/false, a, /*neg_b=*/false, b,
            /*c_mod=*/(short)0, acc, /*reuse_a=*/false, /*reuse_b=*/false);
    }

    // C/D: VGPR r -> M = m0 + half*8 + r, N = n0 + l
    const int mbase = m0 + half * 8;
#pragma unroll
    for (int r = 0; r < 8; ++r) {
        int m = mbase + r;
        if (m < M) C[(long long)m * N + n0 + l] = acc[r];
    }
}

// ------------------------------------------ edge scatter-add (aggregation)
// out[col[e], :] += dis[row[e]]*dis[col[e]] * h[row[e], :]
// One thread per (edge, 4-channel group); float4 gather (global_load_b128),
// fp32 global atomics. h (<=51MB) is L2-resident on MI455X (192MB L2).
template <int LOG2CVEC>  // C/4 = 1<<LOG2CVEC channel groups
__global__ void aggregate_kernel(const float4* __restrict__ h,
                                 const float* __restrict__ dis,
                                 const long long* __restrict__ row,
                                 const long long* __restrict__ col,
                                 float* __restrict__ out, int E) {
    const int cvec = 1 << LOG2CVEC;
    long long idx = (long long)blockIdx.x * blockDim.x + threadIdx.x;
    long long total = (long long)E << LOG2CVEC;
    if (idx >= total) return;
    int e = (int)(idx >> LOG2CVEC);
    int g = (int)(idx & (cvec - 1));
    int r = (int)row[e];
    int c = (int)col[e];
    float norm = dis[r] * dis[c];
    float4 m = h[(long long)r * cvec + g];
    float* o = out + ((long long)c * cvec + g) * 4;
    atomicAdd(o + 0, norm * m.x);
    atomicAdd(o + 1, norm * m.y);
    atomicAdd(o + 2, norm * m.z);
    atomicAdd(o + 3, norm * m.w);
}

// h[i] = relu(h[i] + b[i % C])   (C is a power of two -> mask)
__global__ void bias_relu_kernel(float* __restrict__ h, const float* __restrict__ b,
                                 long long n, int cmask) {
    long long i = (long long)blockIdx.x * blockDim.x + threadIdx.x;
    if (i < n) h[i] = fmaxf(h[i] + b[i & cmask], 0.0f);
}

__global__ void bias_add_kernel(float* __restrict__ o, const float* __restrict__ b,
                                long long n, int cmask) {
    long long i = (long long)blockIdx.x * blockDim.x + threadIdx.x;
    if (i < n) o[i] += b[i & cmask];
}

// ---------------------------------------------------------------- launcher
extern "C" void kernel_launch(void* const* d_in, const int* in_sizes, int n_in,
                              void* d_out, int out_size, void* d_ws, size_t ws_size,
                              hipStream_t stream) {
    (void)in_sizes; (void)n_in; (void)out_size; (void)ws_size;

    const float*     x    = (const float*)d_in[0];
    const long long* ei   = (const long long*)d_in[1];  // int64 edge_index [2, E]
    const float*     W1   = (const float*)d_in[2];
    const float*     b1   = (const float*)d_in[3];
    const float*     W2   = (const float*)d_in[4];
    const float*     b2   = (const float*)d_in[5];
    const long long* row  = ei;             // src
    const long long* col  = ei + N_EDGES;   // dst
    float*           out  = (float*)d_out;

    // workspace layout (floats): dis[N] | hx[N*HID_C] | agg[N*HID_C]
    float* dis = (float*)d_ws;
    float* hx  = dis + N_NODES;                      // x@W1, later h1@W2
    float* agg = hx + (size_t)N_NODES * HID_C;       // layer-1 scatter target

    const long long nH = (long long)N_NODES * HID_C; // 12.8M
    const long long nO = (long long)N_NODES * OUT_C; // 6.4M

    // 1) zero accumulators (required every call)
    zero_kernel<<<1024, 256, 0, stream>>>(dis, N_NODES);
    zero_kernel<<<4096, 256, 0, stream>>>(agg, nH);
    zero_kernel<<<4096, 256, 0, stream>>>(out, nO);

    // 2) degrees -> rsqrt normalizers
    degree_kernel<<<(N_EDGES + 255) / 256, 256, 0, stream>>>(col, dis, N_EDGES);
    rsqrt_kernel<<<(N_NODES + 255) / 256, 256, 0, stream>>>(dis, N_NODES);

    // 3) layer 1: hx = x @ W1   (M=100000, K=64, N=128); 8 waves/block
    gemm_wmma_f32<<<N_NODES / 16, (HID_C / 16) * 32, 0, stream>>>(
        x, W1, hx, N_NODES, IN_C, HID_C);

    // 4) scatter aggregation into agg (C=128 -> 32 float4 groups/edge)
    {
        long long total = (long long)N_EDGES * (HID_C / 4);
        aggregate_kernel<5><<<(unsigned)((total + 255) / 256), 256, 0, stream>>>(
            (const float4*)hx, dis, row, col, agg, N_EDGES);
    }

    // 5) h1 = relu(agg + b1), in place
    bias_relu_kernel<<<(unsigned)((nH + 255) / 256), 256, 0, stream>>>(
        agg, b1, nH, HID_C - 1);

    // 6) layer 2: hx = h1 @ W2  (M=100000, K=128, N=64); 4 waves/block
    gemm_wmma_f32<<<N_NODES / 16, (OUT_C / 16) * 32, 0, stream>>>(
        agg, W2, hx, N_NODES, HID_C, OUT_C);

    // 7) scatter aggregation into out (C=64 -> 16 float4 groups/edge)
    {
        long long total = (long long)N_EDGES * (OUT_C / 4);
        aggregate_kernel<4><<<(unsigned)((total + 255) / 256), 256, 0, stream>>>(
            (const float4*)hx, dis, row, col, out, N_EDGES);
    }

    // 8) out += b2
    bias_add_kernel<<<(unsigned)((nO + 255) / 256), 256, 0, stream>>>(
        out, b2, nO, OUT_C - 1);
}